// RNade_9740985827985
// MI455X (gfx1250) — compile-verified
//
#include <hip/hip_runtime.h>

// Problem constants (from reference):
#define B_TOT   256
#define L_TOT   500
#define D_IN    3
#define C_IN    32
#define K_IN    35      // D+C
#define K_PAD   36      // padded to multiple of 4 for 16x16x4 WMMA
#define H_DIM   512
#define O_DIM   15
#define M_DIM   5
#define OUT_DIM 35      // 2*O + M
#define OUT_PAD 48      // padded to 3 N-tiles of 16

// Chunked-scan configuration
#define CL      20                  // steps per chunk
#define NCH     25                  // 25*20 = 500
#define BT      16                  // batch tile (WMMA M)
#define NBT     (B_TOT / BT)        // 16 batch tiles
#define WAVES   8
#define TPB     (WAVES * 32)        // 256 threads, 8 wave32
#define WCOLS   64                  // H columns per wave (512/8)
#define NT_ENC  4                   // 64/16 N-tiles per wave (encoder)
#define NT_OUT  3                   // 48/16 N-tiles (decoder)
#define KS_OUT  16                  // 64/4 K-steps (decoder, per-wave K range)
#define HPAD    17                  // LDS row padding (bank-conflict free)

typedef __attribute__((ext_vector_type(2))) float v2f;
typedef __attribute__((ext_vector_type(8))) float v8f;

__device__ __forceinline__ v8f wmma_f32(v2f a, v2f b, v8f c) {
  // V_WMMA_F32_16X16X4_F32: D = A(16x4) * B(4x16) + C(16x16), fp32 throughout
  return __builtin_amdgcn_wmma_f32_16x16x4_f32(
      /*neg_a=*/false, a, /*neg_b=*/false, b,
      /*c_mod=*/(short)0, c, /*reuse_a=*/false, /*reuse_b=*/false);
}

// ---------------------------------------------------------------------------
// Stage xz_l into LDS as [kk][b] (kk = 0..35, row 35 zero) with fully
// uniform, unconditional passes (no loop-carried exec divergence).
// ---------------------------------------------------------------------------
__device__ __forceinline__ void stage_xz(const float* __restrict__ x,
                                         const float* __restrict__ z,
                                         int bt, int l, int tid,
                                         float* __restrict__ ldsXZ) {
  // z rows kk = 3..34: exactly 512 elements = 2 full uniform passes
  {
    int e = tid;                                   // pass 0: kk = 3..18
    int kk = D_IN + (e >> 4), b = e & 15;
    int gb = bt * BT + b;
    ldsXZ[kk * BT + b] = z[((size_t)gb * L_TOT + l) * C_IN + (kk - D_IN)];
    e = tid + TPB;                                 // pass 1: kk = 19..34
    kk = D_IN + (e >> 4); b = e & 15;
    gb = bt * BT + b;
    ldsXZ[kk * BT + b] = z[((size_t)gb * L_TOT + l) * C_IN + (kk - D_IN)];
  }
  if (tid < D_IN * BT) {                           // x rows kk = 0..2 (48 elems)
    int kk = tid >> 4, b = tid & 15;
    int gb = bt * BT + b;
    ldsXZ[kk * BT + b] = x[((size_t)gb * L_TOT + l) * D_IN + kk];
  }
  if (tid < BT) ldsXZ[K_IN * BT + tid] = 0.0f;     // zero-pad row kk = 35
}

// ---------------------------------------------------------------------------
// Encoder WMMA chain: acc += xz_l @ W_l for this wave's 64 H columns.
// 8 unconditional K-steps + peeled tail (only kk==35 masked; its A column
// is zero in LDS so the product is exact).
// ---------------------------------------------------------------------------
__device__ __forceinline__ void enc_step(const float* __restrict__ Wl,
                                         const float* __restrict__ ldsXZ,
                                         int hi, int q16, int colbase,
                                         v8f acc[NT_ENC]) {
#pragma unroll 4
  for (int k = 0; k < 8; ++k) {
    const int k0 = 4 * k + 2 * hi;                 // frag reg j holds K = 4k+j+2*hi
    v2f a;
    a.x = ldsXZ[(k0    ) * BT + q16];
    a.y = ldsXZ[(k0 + 1) * BT + q16];
#pragma unroll
    for (int t = 0; t < NT_ENC; ++t) {
      const int col = colbase + 16 * t + q16;
      v2f b;
      b.x = Wl[(size_t)(k0    ) * H_DIM + col];
      b.y = Wl[(size_t)(k0 + 1) * H_DIM + col];
      acc[t] = wmma_f32(a, b, acc[t]);
    }
  }
  {                                                // tail: kk = 32 + 2*hi (+1)
    const int k0 = 32 + 2 * hi;
    v2f a;
    a.x = ldsXZ[(k0    ) * BT + q16];
    a.y = ldsXZ[(k0 + 1) * BT + q16];              // row 35 reads 0 from LDS
    const bool oky = (k0 + 1 < K_IN);              // hi==1 -> kk=35 out of W bounds
#pragma unroll
    for (int t = 0; t < NT_ENC; ++t) {
      const int col = colbase + 16 * t + q16;
      v2f b;
      b.x = Wl[(size_t)(k0) * H_DIM + col];
      b.y = oky ? Wl[(size_t)(k0 + 1) * H_DIM + col] : 0.0f;
      acc[t] = wmma_f32(a, b, acc[t]);
    }
  }
}

// ---------------------------------------------------------------------------
// Phase 1: per-chunk sums S[ch][b][h] = sum_{l in chunk} m_l[b][h]
// ---------------------------------------------------------------------------
__global__ __launch_bounds__(TPB)
void rnade_chunk_sums(const float* __restrict__ x, const float* __restrict__ z,
                      const float* __restrict__ W, float* __restrict__ wsS) {
  __shared__ float ldsXZ[K_PAD * BT];

  const int ch   = blockIdx.x;
  const int bt   = blockIdx.y;
  const int tid  = threadIdx.x;
  const int lane = tid & 31;
  const int w    = tid >> 5;
  const int q16  = lane & 15;
  const int hi   = lane >> 4;
  const int colbase = w * WCOLS;

  v8f acc[NT_ENC] = {};                // running sum of m over the chunk

  for (int li = 0; li < CL; ++li) {
    const int l = ch * CL + li;
    const float* Wl = W + (size_t)l * K_IN * H_DIM;
    __builtin_prefetch(Wl + K_IN * H_DIM);         // global_prefetch_b8

    __syncthreads();                               // protect ldsXZ reuse
    stage_xz(x, z, bt, l, tid, ldsXZ);
    __syncthreads();

    enc_step(Wl, ldsXZ, hi, q16, colbase, acc);
  }

  // store chunk sum: C layout -> ws[ch][b][h]
#pragma unroll
  for (int t = 0; t < NT_ENC; ++t) {
    const int h = colbase + 16 * t + q16;
#pragma unroll
    for (int g = 0; g < 8; ++g) {
      const int b = bt * BT + g + hi * 8;
      wsS[((size_t)ch * B_TOT + b) * H_DIM + h] = acc[t][g];
    }
  }
}

// ---------------------------------------------------------------------------
// Phase 2: in-place exclusive prefix over chunks, per (b,h)
// ---------------------------------------------------------------------------
__global__ __launch_bounds__(256)
void rnade_prefix(float* __restrict__ wsS) {
  const int gid = blockIdx.x * 256 + threadIdx.x;  // gid = b*H + h
  if (gid >= B_TOT * H_DIM) return;
  float run = 0.0f;
  for (int ch = 0; ch < NCH; ++ch) {
    const size_t idx = (size_t)ch * B_TOT * H_DIM + gid;
    const float v = wsS[idx];
    wsS[idx] = run;
    run += v;
  }
}

// ---------------------------------------------------------------------------
// Phase 3: recompute m within chunk, produce h and decoder outputs
// ---------------------------------------------------------------------------
__global__ __launch_bounds__(TPB)
void rnade_output(const float* __restrict__ x, const float* __restrict__ z,
                  const float* __restrict__ W, const float* __restrict__ b_enc,
                  const float* __restrict__ V_mu, const float* __restrict__ b_mu,
                  const float* __restrict__ V_sigma, const float* __restrict__ b_sigma,
                  const float* __restrict__ V_pi, const float* __restrict__ b_pi,
                  const float* __restrict__ wsS, float* __restrict__ out) {
  __shared__ float ldsXZ[K_PAD * BT];        // staged inputs for one step
  __shared__ float ldsH[H_DIM * HPAD];       // h transposed: [h][b], stride 17
  __shared__ float ldsO[OUT_PAD * HPAD];     // output accum: [o][b], stride 17

  const int ch   = blockIdx.x;
  const int bt   = blockIdx.y;
  const int tid  = threadIdx.x;
  const int lane = tid & 31;
  const int w    = tid >> 5;
  const int q16  = lane & 15;
  const int hi   = lane >> 4;
  const int colbase = w * WCOLS;

  // encoder bias for this wave's columns (depends on h only)
  float be[NT_ENC];
#pragma unroll
  for (int t = 0; t < NT_ENC; ++t) be[t] = b_enc[colbase + 16 * t + q16];

  // init running accumulator from exclusive chunk prefix P[ch]
  v8f acc[NT_ENC];
#pragma unroll
  for (int t = 0; t < NT_ENC; ++t) {
    const int h = colbase + 16 * t + q16;
#pragma unroll
    for (int g = 0; g < 8; ++g) {
      const int b = bt * BT + g + hi * 8;
      acc[t][g] = wsS[((size_t)ch * B_TOT + b) * H_DIM + h];
    }
  }

  for (int li = 0; li < CL; ++li) {
    const int l = ch * CL + li;
    const float rl = (l > 0) ? (1.0f / (float)l) : 1.0f;
    const float* Wl  = W       + (size_t)l * K_IN * H_DIM;
    const float* Vmu = V_mu    + (size_t)l * H_DIM * O_DIM;
    const float* Vsg = V_sigma + (size_t)l * H_DIM * O_DIM;
    const float* Vpi = V_pi    + (size_t)l * H_DIM * M_DIM;
    __builtin_prefetch(Wl + K_IN * H_DIM);

    // per-lane decoder weight pointer/stride (branchless gathers in hot loop;
    // padded lanes o>=35 read a dummy value into discarded LDS columns)
    const float* vb[NT_OUT];
    int vs[NT_OUT];
#pragma unroll
    for (int n = 0; n < NT_OUT; ++n) {
      const int o = 16 * n + q16;
      const float* p = Vmu + o;          int s = O_DIM;
      if (o >= O_DIM)     { p = Vsg + (o - O_DIM);     s = O_DIM; }
      if (o >= 2 * O_DIM) { p = Vpi + (o - 2 * O_DIM); s = M_DIM; }
      if (o >= OUT_DIM)   { p = Vmu;                   s = 0;     }
      vb[n] = p; vs[n] = s;
    }

    __syncthreads();   // prev-iter readers of ldsXZ/ldsO done

    stage_xz(x, z, bt, l, tid, ldsXZ);

    // init output accumulator with decoder biases: 768 = 3 uniform passes
#pragma unroll
    for (int rep = 0; rep < 3; ++rep) {
      const int e = tid + rep * TPB;
      const int o = e >> 4, b = e & 15;
      const float* p = b_mu + (size_t)l * O_DIM + o;
      if (o >= O_DIM)     p = b_sigma + (size_t)l * O_DIM + (o - O_DIM);
      if (o >= 2 * O_DIM) p = b_pi + (size_t)l * M_DIM + (o - 2 * O_DIM);
      if (o >= OUT_DIM)   p = b_mu;      // dummy for padded cols (discarded)
      ldsO[o * HPAD + b] = *p;
    }

    // h_l = relu(r_l * (b_enc + acc))  -- acc is the EXCLUSIVE prefix here.
    // Store transposed [h][b] into this wave's own LDS region.
#pragma unroll
    for (int t = 0; t < NT_ENC; ++t) {
      const int hcol = colbase + 16 * t + q16;
#pragma unroll
      for (int g = 0; g < 8; ++g) {
        const float av = rl * (be[t] + acc[t][g]);
        ldsH[hcol * HPAD + g + hi * 8] = fmaxf(av, 0.0f);
      }
    }

    __syncthreads();   // ldsXZ + ldsO visible to all waves

    // encoder: acc += xz_l @ W_l (scan state updated AFTER h was taken)
    enc_step(Wl, ldsXZ, hi, q16, colbase, acc);

    // decoder: partial over this wave's 64 h-columns (A from LDS, M=batch)
    v8f op[NT_OUT] = {};
#pragma unroll 4
    for (int k = 0; k < KS_OUT; ++k) {
      const int kk0 = colbase + 4 * k + 2 * hi;    // absolute h index
      v2f a;
      a.x = ldsH[(kk0    ) * HPAD + q16];
      a.y = ldsH[(kk0 + 1) * HPAD + q16];
#pragma unroll
      for (int n = 0; n < NT_OUT; ++n) {
        v2f b;
        b.x = vb[n][(size_t)(kk0    ) * vs[n]];
        b.y = vb[n][(size_t)(kk0 + 1) * vs[n]];
        op[n] = wmma_f32(a, b, op[n]);
      }
    }

    // cross-wave reduction of decoder partials via LDS float atomics
#pragma unroll
    for (int n = 0; n < NT_OUT; ++n) {
      const int o = 16 * n + q16;
#pragma unroll
      for (int g = 0; g < 8; ++g) {
        atomicAdd(&ldsO[o * HPAD + g + hi * 8], op[n][g]);
      }
    }
    __syncthreads();   // all partials accumulated

    // write out[b][l][o]: 560 elements = 2 full passes + 48
#pragma unroll
    for (int rep = 0; rep < 3; ++rep) {
      const int e = tid + rep * TPB;
      if (e < BT * OUT_DIM) {
        const int b = e / OUT_DIM, o = e - b * OUT_DIM;
        out[(((size_t)(bt * BT + b)) * L_TOT + l) * OUT_DIM + o] = ldsO[o * HPAD + b];
      }
    }
  }
}

// ---------------------------------------------------------------------------
extern "C" void kernel_launch(void* const* d_in, const int* in_sizes, int n_in,
                              void* d_out, int out_size, void* d_ws, size_t ws_size,
                              hipStream_t stream) {
  const float* x       = (const float*)d_in[0];
  const float* z       = (const float*)d_in[1];
  const float* W_enc   = (const float*)d_in[2];
  const float* b_enc   = (const float*)d_in[3];
  const float* V_mu    = (const float*)d_in[4];
  const float* b_mu    = (const float*)d_in[5];
  const float* V_sigma = (const float*)d_in[6];
  const float* b_sigma = (const float*)d_in[7];
  const float* V_pi    = (const float*)d_in[8];
  const float* b_pi    = (const float*)d_in[9];
  float* out = (float*)d_out;
  float* wsS = (float*)d_ws;   // NCH*B*H floats = 13.1 MB chunk sums / prefixes

  dim3 grid(NCH, NBT);
  rnade_chunk_sums<<<grid, TPB, 0, stream>>>(x, z, W_enc, wsS);
  rnade_prefix<<<(B_TOT * H_DIM) / 256, 256, 0, stream>>>(wsS);
  rnade_output<<<grid, TPB, 0, stream>>>(x, z, W_enc, b_enc, V_mu, b_mu,
                                         V_sigma, b_sigma, V_pi, b_pi, wsS, out);
}